// Predictor_34737695490211
// MI455X (gfx1250) — compile-verified
//
#include <hip/hip_runtime.h>
#include <hip/hip_bf16.h>

// MI455X (gfx1250, wave32). This workload is HBM-bound (~88MB fp32 weights,
// ~0.67 GFLOP -> 7.6 FLOP/B): the kernels below stream skip_filter /
// dense_kernels / conv_filters exactly once and do the skinny GEMMs with
// V_WMMA_F32_16X16X4_F32 (M=16 == batch), deferring all per-sample 1/norm
// scalars to a final epilogue (the computation is linear in h).

typedef __attribute__((ext_vector_type(2))) float v2f;
typedef __attribute__((ext_vector_type(8))) float v8f;

#define BB   16
#define FF   2048
#define TT   256
#define NDEP 4
#define EPSV 1e-8f

// ---------------------------------------------------------------------------
// modp1[i][b,f] = t_global[b,:] @ dense_kernels[i][:,f] + bias[i][f] + 1
// One wave per 16-wide f tile. grid = 4 * 128 blocks of 32 threads.
// WMMA: D[M=b][N=f] = A[b][t] * B[t][f], K=256 in steps of 4.
// ---------------------------------------------------------------------------
__global__ void mod_kernel(const float* __restrict__ tg,
                           const float* __restrict__ dk,
                           const float* __restrict__ bias,
                           float* __restrict__ modp1) {
  const int depth = blockIdx.x >> 7;
  const int f0    = (blockIdx.x & 127) << 4;
  const int lane  = threadIdx.x;
  const int lo    = lane & 15;
  const int hi    = lane >> 4;
  const float* dkd = dk + (size_t)depth * (TT * FF);
  v8f c = {};
  for (int k = 0; k < TT; k += 4) {
    // A frag (16x4 f32): lane<16 -> A[M=lo][K=k..k+1], lane>=16 -> K=k+2..k+3
    v2f a = *(const v2f*)(tg + lo * TT + k + 2 * hi);
    // B frag (4x16 f32): VGPR0 = rows K=k|k+2, VGPR1 = rows K=k+1|k+3, N=f0+lo
    v2f b;
    b.x = dkd[(size_t)(k + 2 * hi + 0) * FF + f0 + lo];
    b.y = dkd[(size_t)(k + 2 * hi + 1) * FF + f0 + lo];
    c = __builtin_amdgcn_wmma_f32_16x16x4_f32(false, a, false, b, (short)0, c,
                                              false, false);
  }
  const float* bi = bias + depth * FF;
  float* mo = modp1 + (size_t)depth * (BB * FF);
  float add = bi[f0 + lo] + 1.0f;
#pragma unroll
  for (int j = 0; j < 8; ++j) {
    int bM = j + 8 * hi;                       // D: VGPR j -> M=j (lanes 0-15), M=j+8 (16-31)
    mo[(size_t)bM * FF + f0 + lo] = c[j] + add;
  }
}

// ---------------------------------------------------------------------------
// Skip path collapses to a column sum: part[g][c] = sum over 32 rows of tile g.
// ---------------------------------------------------------------------------
__global__ void colsum_part_kernel(const float* __restrict__ skip,
                                   float* __restrict__ part) {
  const int g = blockIdx.x;  // 64 tiles of 32 rows
  for (int cc = threadIdx.x; cc < FF; cc += 256) {
    float s = 0.f;
#pragma unroll
    for (int r = 0; r < 32; ++r) s += skip[(size_t)(g * 32 + r) * FF + cc];
    part[(size_t)g * FF + cc] = s;
  }
}

__global__ void colsum_reduce_kernel(const float* __restrict__ part,
                                     float* __restrict__ colsum) {
  const int cc = blockIdx.x * 256 + threadIdx.x;  // 8 blocks x 256
  float s = 0.f;
  for (int g = 0; g < 64; ++g) s += part[(size_t)g * FF + cc];
  colsum[cc] = s;
}

// ---------------------------------------------------------------------------
// One depth stage, norms deferred:
//   hout[b,f] = modp1[b,f] * sum_c conv[f,c]*hin[b,c]
//   nspart[blk][b] = sum_{f in tile} modp1[b,f]^2 * rowsq[f]
// grid = 128 f-tiles, block = 256 (8 waves, K split 256 each).
// WMMA: D[M=f_local][N=b] = A[f][c] * B[c][b],  B[c][b] = hin[b][c].
// rowsq computed for free from the streamed A fragments.
// ---------------------------------------------------------------------------
__global__ void depth_kernel(const float* __restrict__ conv,
                             const float* __restrict__ hin,
                             const float* __restrict__ modp1,
                             float* __restrict__ hout,
                             float* __restrict__ nspart) {
  __shared__ float cred[8 * 256];
  __shared__ float sqred[8 * 32];
  __shared__ float rowsq_s[16];
  const int f0   = blockIdx.x << 4;
  const int wave = threadIdx.x >> 5;
  const int lane = threadIdx.x & 31;
  const int lo   = lane & 15;
  const int hi   = lane >> 4;
  const float* crow = conv + (size_t)(f0 + lo) * FF;  // conv row for this lane
  const float* hrow = hin + (size_t)lo * FF;          // sample row for B frag
  v8f c = {};
  float sq = 0.f;
  const int k0 = wave * 256;
  for (int k = k0; k < k0 + 256; k += 4) {
    v2f a = *(const v2f*)(crow + k + 2 * hi);  // A[f=f0+lo][c=k(+2hi)..]
    v2f b = *(const v2f*)(hrow + k + 2 * hi);  // B[c][b=lo] = hin[lo][c]
    sq = fmaf(a.x, a.x, sq);
    sq = fmaf(a.y, a.y, sq);
    c = __builtin_amdgcn_wmma_f32_16x16x4_f32(false, a, false, b, (short)0, c,
                                              false, false);
  }
#pragma unroll
  for (int j = 0; j < 8; ++j)
    cred[wave * 256 + (j + 8 * hi) * 16 + lo] = c[j];  // idx = f_local*16 + b
  sqred[wave * 32 + lane] = sq;
  __syncthreads();
  {  // reduce 8 K-partials; apply (mod+1); store unnormalized hout
    const int t = threadIdx.x;
    float y = 0.f;
#pragma unroll
    for (int w = 0; w < 8; ++w) y += cred[w * 256 + t];
    const int fl = t >> 4, bb = t & 15;
    const float m = modp1[(size_t)bb * FF + f0 + fl];
    hout[(size_t)bb * FF + f0 + fl] = y * m;
  }
  if (threadIdx.x < 16) {  // rowsq[f] = lane lo + lane lo+16 partials, all waves
    float s = 0.f;
#pragma unroll
    for (int w = 0; w < 8; ++w)
      s += sqred[w * 32 + threadIdx.x] + sqred[w * 32 + 16 + threadIdx.x];
    rowsq_s[threadIdx.x] = s;
  }
  __syncthreads();
  if (threadIdx.x < 16) {  // per-sample normsq partial for this f tile
    const int bb = threadIdx.x;
    float acc = 0.f;
#pragma unroll
    for (int fl = 0; fl < 16; ++fl) {
      const float m = modp1[(size_t)bb * FF + f0 + fl];
      acc = fmaf(m * m, rowsq_s[fl], acc);
    }
    nspart[(size_t)blockIdx.x * 16 + bb] = acc;
  }
}

// ---------------------------------------------------------------------------
// out[b] = dot(x[b], colsum) + (prod_i 1/sqrt(max(normsq_i[b],EPS))) * sum_f h4[b,f]
// ---------------------------------------------------------------------------
__global__ void final_kernel(const float* __restrict__ x,
                             const float* __restrict__ h4,
                             const float* __restrict__ colsum,
                             const float* __restrict__ nspart,  // [4][128][16]
                             float* __restrict__ out) {
  __shared__ float ns[NDEP * 16];
  __shared__ float red[256];
  const int t = threadIdx.x;
  if (t < NDEP * 16) {
    const int d = t >> 4, bb = t & 15;
    float s = 0.f;
    for (int g = 0; g < 128; ++g) s += nspart[(size_t)(d * 128 + g) * 16 + bb];
    ns[t] = s;
  }
  __syncthreads();
  const int bb = t >> 4;  // 16 threads per sample
  const int k  = t & 15;
  float s_skip = 0.f, s_h = 0.f;
  for (int cc = k; cc < FF; cc += 16) {
    s_skip = fmaf(x[(size_t)bb * FF + cc], colsum[cc], s_skip);
    s_h += h4[(size_t)bb * FF + cc];
  }
  float sprod = 1.f;
#pragma unroll
  for (int i = 0; i < NDEP; ++i) sprod /= sqrtf(fmaxf(ns[i * 16 + bb], EPSV));
  red[t] = s_skip + s_h * sprod;
  __syncthreads();
#pragma unroll
  for (int off = 8; off > 0; off >>= 1) {
    if (k < off) red[t] += red[t + off];
    __syncthreads();
  }
  if (k == 0) out[bb] = red[t];
}

// ---------------------------------------------------------------------------
extern "C" void kernel_launch(void* const* d_in, const int* in_sizes, int n_in,
                              void* d_out, int out_size, void* d_ws,
                              size_t ws_size, hipStream_t stream) {
  (void)in_sizes; (void)n_in; (void)out_size; (void)ws_size;
  const float* x     = (const float*)d_in[0];  // (16,32,8,8) == (16,2048)
  const float* tg    = (const float*)d_in[1];  // (16,256)
  const float* skipf = (const float*)d_in[2];  // (2048,2048)
  const float* dk    = (const float*)d_in[3];  // (4,256,2048)
  const float* db    = (const float*)d_in[4];  // (4,2048)
  const float* cf    = (const float*)d_in[5];  // (4,2048,2048)
  float* out = (float*)d_out;

  float* ws     = (float*)d_ws;
  float* modp1  = ws;                                  // 4*16*2048 = 131072
  float* hA     = modp1 + NDEP * BB * FF;              // 16*2048
  float* hB     = hA + BB * FF;                        // 16*2048
  float* cspart = hB + BB * FF;                        // 64*2048  = 131072
  float* colsum = cspart + 64 * FF;                    // 2048
  float* nspart = colsum + FF;                         // 4*128*16 = 8192

  mod_kernel<<<NDEP * 128, 32, 0, stream>>>(tg, dk, db, modp1);
  colsum_part_kernel<<<64, 256, 0, stream>>>(skipf, cspart);
  colsum_reduce_kernel<<<8, 256, 0, stream>>>(cspart, colsum);

  const float* hin = x;
  float* houts[NDEP] = {hA, hB, hA, hB};
  for (int i = 0; i < NDEP; ++i) {
    depth_kernel<<<128, 256, 0, stream>>>(cf + (size_t)i * FF * FF, hin,
                                          modp1 + (size_t)i * BB * FF,
                                          houts[i],
                                          nspart + (size_t)i * 128 * 16);
    hin = houts[i];
  }
  final_kernel<<<1, 256, 0, stream>>>(x, hin, colsum, nspart, out);
}